// ZDecoder_23295902614240
// MI455X (gfx1250) — compile-verified
//
#include <hip/hip_runtime.h>
#include <hip/hip_bf16.h>

typedef __attribute__((ext_vector_type(2))) float v2f;
typedef __attribute__((ext_vector_type(4))) float v4f;
typedef __attribute__((ext_vector_type(8))) float v8f;

#define B_   32
#define L_   6
#define R_   8
#define H_   64
#define NPOS (B_ * L_ * R_)          // 1536
#define RPL  262144                  // 8^6
// output flat size = B_ * RPL * L_ = 50331648 floats; 12 floats / thread

// ---------------------------------------------------------------------------
// Phase 1: q0[b][l][r] = W3[0] . relu(W2 . relu(W1 . [rp, phi] + b1) + b2) + b3[0]
// One wave (32 threads) per 16-position M-tile; 64x64 layer via
// V_WMMA_F32_16X16X4_F32 (4 N-tiles x 16 K-steps).
// ---------------------------------------------------------------------------
__global__ __launch_bounds__(32) void zdec_mlp_kernel(
    const float* __restrict__ phi,   // (32,6)
    const float* __restrict__ rp,    // (6,8,1)
    const float* __restrict__ W1,    // (64,2)
    const float* __restrict__ b1,    // (64)
    const float* __restrict__ W2,    // (64,64)
    const float* __restrict__ b2,    // (64)
    const float* __restrict__ W3,    // (6,64) -- only row 0 used
    const float* __restrict__ b3,    // (6)    -- only b3[0] used
    float* __restrict__ q0)          // (32,6,8) workspace
{
    __shared__ float h1[16 * H_];    // 16 positions x 64 hidden
    __shared__ float red[8 * 32];

    const int lane = threadIdx.x;    // 0..31 (wave32)
    const int m    = lane & 15;      // row within M-tile
    const int pos0 = blockIdx.x * 16;

    // ---- layer 1: h1[m][h] = relu(x*W1[h][0] + p*W1[h][1] + b1[h]) ----
    {
        int pos = pos0 + m;
        int b = pos / (L_ * R_);
        int l = (pos / R_) % L_;
        int r = pos % R_;
        float x = rp[l * R_ + r];        // Z == 1
        float p = phi[b * L_ + l];
        int h0 = (lane >> 4) * 32;       // lanes 0-15: h=0..31, lanes 16-31: h=32..63
#pragma unroll
        for (int hh = 0; hh < 32; ++hh) {
            int h = h0 + hh;
            float v = fmaf(x, W1[h * 2 + 0], fmaf(p, W1[h * 2 + 1], b1[h]));
            h1[m * H_ + h] = v > 0.f ? v : 0.f;
        }
    }
    __syncthreads();

    // ---- layer 2 via WMMA + fused layer 3 partial products ----
    // A (16x4 f32) per ISA layout: lane m=lane&15, VGPR0/1 = K + (lane>=16 ? 2 : 0)
    const int koff = (lane >> 4) * 2;

    float acc[8];
#pragma unroll
    for (int j = 0; j < 8; ++j) acc[j] = 0.f;

#pragma unroll
    for (int nt = 0; nt < 4; ++nt) {
        const int n = nt * 16 + m;       // output channel handled by this lane's column
        const float bn = b2[n];
        v8f C;
#pragma unroll
        for (int j = 0; j < 8; ++j) C[j] = bn;   // seed accumulator with bias

#pragma unroll
        for (int k = 0; k < 16; ++k) {           // K = 64 in steps of 4
            v2f A, Bm;
            A[0]  = h1[m * H_ + k * 4 + koff];
            A[1]  = h1[m * H_ + k * 4 + koff + 1];
            // B[K][N] = W2^T -> B element (kk, n) = W2[n*64 + kk]
            Bm[0] = W2[n * H_ + k * 4 + koff];
            Bm[1] = W2[n * H_ + k * 4 + koff + 1];
            C = __builtin_amdgcn_wmma_f32_16x16x4_f32(
                    /*neg_a=*/false, A, /*neg_b=*/false, Bm,
                    /*c_mod=*/(short)0, C, /*reuse_a=*/false, /*reuse_b=*/false);
        }

        // relu + multiply by W3 row 0, accumulate layer-3 partials per lane
        const float w3n = W3[n];
#pragma unroll
        for (int j = 0; j < 8; ++j) {
            float h2 = C[j] > 0.f ? C[j] : 0.f;
            acc[j] = fmaf(h2, w3n, acc[j]);
        }
    }

    // C/D layout: VGPR j holds row (j + 8*(lane>=16)), col = lane&15.
    // Reduce the 16 columns of each row half via LDS.
#pragma unroll
    for (int j = 0; j < 8; ++j) red[j * 32 + lane] = acc[j];
    __syncthreads();

    if (lane < 16) {
        float s = b3[0];
        int base = (lane & 7) * 32 + (lane >> 3) * 16;
#pragma unroll
        for (int t = 0; t < 16; ++t) s += red[base + t];
        q0[pos0 + lane] = s;
    }
}

// ---------------------------------------------------------------------------
// Phase 2: out[b][n][l] = q0[b][l][(n >> 3l) & 7]  -- pure store bandwidth.
// 12 contiguous floats (2 combos) per thread, 3x nontemporal b128 stores.
// ---------------------------------------------------------------------------
__global__ __launch_bounds__(256) void zdec_bcast_kernel(
    const float* __restrict__ q0, float* __restrict__ out, int nthreads)
{
    __shared__ float sq[NPOS];
    for (int i = threadIdx.x; i < NPOS; i += 256) sq[i] = q0[i];
    __syncthreads();

    int gtid = blockIdx.x * 256 + threadIdx.x;
    if (gtid >= nthreads) return;

    // per-b span = RPL * 6 = 12 * 2^17 floats -> b = gtid >> 17, n0 = (gtid & 131071)*2
    const int b    = gtid >> 17;
    const int n0   = (gtid & 131071) << 1;
    const int base = gtid * 12;          // 16B-aligned float offset
    const float* qb = &sq[b * (L_ * R_)];

    float vals[12];
#pragma unroll
    for (int c = 0; c < 2; ++c) {
        int n = n0 + c;
#pragma unroll
        for (int l = 0; l < L_; ++l) {
            vals[c * 6 + l] = qb[l * R_ + ((n >> (3 * l)) & 7)];
        }
    }

    v4f* o = (v4f*)(out + base);
    v4f v0 = { vals[0], vals[1], vals[2],  vals[3]  };
    v4f v1 = { vals[4], vals[5], vals[6],  vals[7]  };
    v4f v2 = { vals[8], vals[9], vals[10], vals[11] };
    __builtin_nontemporal_store(v0, o + 0);
    __builtin_nontemporal_store(v1, o + 1);
    __builtin_nontemporal_store(v2, o + 2);
}

// ---------------------------------------------------------------------------
extern "C" void kernel_launch(void* const* d_in, const int* in_sizes, int n_in,
                              void* d_out, int out_size, void* d_ws, size_t ws_size,
                              hipStream_t stream) {
    (void)in_sizes; (void)n_in; (void)ws_size;

    const float* phi = (const float*)d_in[0];
    const float* rp  = (const float*)d_in[1];
    const float* W1  = (const float*)d_in[2];
    const float* b1  = (const float*)d_in[3];
    const float* W2  = (const float*)d_in[4];
    const float* b2  = (const float*)d_in[5];
    const float* W3  = (const float*)d_in[6];
    const float* b3  = (const float*)d_in[7];

    float* out = (float*)d_out;
    float* q0  = (float*)d_ws;           // 1536 floats of scratch

    // Phase 1: 96 M-tiles of 16 positions, one wave each.
    zdec_mlp_kernel<<<NPOS / 16, 32, 0, stream>>>(phi, rp, W1, b1, W2, b2, W3, b3, q0);

    // Phase 2: 12 floats per thread over the whole output.
    int nthreads = (out_size + 11) / 12;             // 4,194,304 for the fixed shapes
    int blocks   = (nthreads + 255) / 256;
    zdec_bcast_kernel<<<blocks, 256, 0, stream>>>(q0, out, nthreads);
}